// AtomDistances_7602092114189
// MI455X (gfx1250) — compile-verified
//
#include <hip/hip_runtime.h>
#include <hip/hip_bf16.h>

typedef float v2f __attribute__((ext_vector_type(2)));
typedef float v8f __attribute__((ext_vector_type(8)));

#define B_   16
#define A_   4096
#define N_   128
#define PB   (A_ * N_)        // pairs per batch = 524288 (divisible by 32)
#define TOTAL ((long long)B_ * PB)

// dist[b,a,n] = || pos[b,nbr]-pos[b,a] + cell_offsets[b,a,n,:] @ cell[b] || * mask
//
// WMMA mapping (V_WMMA_F32_16X16X4_F32, D = A*B + C):
//   A (16x4, M x K): A[m][k] = cell[k][m]  (m<3, k<3; rest 0)  -> A = cell^T
//   B (4x16, K x N): B[k][n] = cell_offsets[pair n][k]
//   C (16x16)      : C[e][n] = pos_diff_e[pair n]  (e<3; rest 0)
//   D[e][n] = dist_vec_e[pair n]   -> lane n (0..15) owns D[0..2] = (x,y,z)
__global__ __launch_bounds__(256) void AtomDistances_kernel(
    const float* __restrict__ positions,        // (B, A, 3) f32
    const long long* __restrict__ neighbors,    // (B, A, N) i64
    const float* __restrict__ cell,             // (B, 3, 3) f32
    const float* __restrict__ cell_offsets,     // (B, A, N, 3) f32
    const unsigned char* __restrict__ mask,     // (B, A, N) bool
    float* __restrict__ out)                    // (B, A, N) f32
{
    const int t   = threadIdx.x & 31;                       // lane id (wave32)
    const int gw  = (int)((blockIdx.x * blockDim.x + threadIdx.x) >> 5);
    const int nw  = (int)((gridDim.x * blockDim.x) >> 5);
    const bool lo = (t < 16);
    const int  m  = t & 15;

    for (long long pbase = (long long)gw * 32; pbase < TOTAL; pbase += (long long)nw * 32) {
        const int b      = (int)(pbase / PB);
        const int pLocal = (int)(pbase % PB);
        const int a      = pLocal >> 7;                     // uniform: 32-chunk never crosses an atom
        const long long gp = pbase + t;                     // this lane's global pair id

        const float* cb   = cell + (size_t)b * 9;           // cell[d][e] = cb[3*d + e]
        const float* posb = positions + (size_t)b * A_ * 3;

        // ---- streaming loads (non-temporal: zero reuse, keep L2 for positions) ----
        const int nbr = (int)__builtin_nontemporal_load(&neighbors[gp]);
        const float* cop = cell_offsets + (size_t)gp * 3;
        const float ox = __builtin_nontemporal_load(&cop[0]);
        const float oy = __builtin_nontemporal_load(&cop[1]);
        const float oz = __builtin_nontemporal_load(&cop[2]);
        const unsigned char mk = __builtin_nontemporal_load(&mask[gp]);

        // ---- cached gathers (48 KB/batch lives in L2) ----
        const float* np = posb + (size_t)nbr * 3;
        const float* ap = posb + (size_t)a * 3;
        const float dx = np[0] - ap[0];
        const float dy = np[1] - ap[1];
        const float dz = np[2] - ap[2];

        // ---- A = cell^T : lanes 0-15 hold K0/K1, lanes 16-31 hold K2/K3 ----
        const float a0 = (m < 3) ? (lo ? cb[m] : cb[6 + m]) : 0.0f;   // K0: cell[0][m], K2: cell[2][m]
        const float a1 = (lo && m < 3) ? cb[3 + m] : 0.0f;            // K1: cell[1][m], K3: 0
        const v2f Am = {a0, a1};

        // ---- cross-half shuffles to build B / C half-wave layouts ----
        const float ozp = __shfl(oz, t - 16, 32);   // pair (t-16) z  -> upper half of B1
        const float oxp = __shfl(ox, t + 16, 32);   // pair (t+16) x  -> lower half of B2
        const float oyp = __shfl(oy, t + 16, 32);
        const float dxp = __shfl(dx, t + 16, 32);
        const float dyp = __shfl(dy, t + 16, 32);
        const float dzp = __shfl(dz, t + 16, 32);

        // B (4x16): VGPR0 = {K0 | K2}, VGPR1 = {K1 | K3} across the two half-waves
        const v2f B1 = { lo ? ox  : ozp, lo ? oy  : 0.0f };   // pairs pbase+0..15
        const v2f B2 = { lo ? oxp : oz,  lo ? oyp : 0.0f };   // pairs pbase+16..31

        // C: rows 0..2 = pos_diff components, rest 0 (upper half-wave = rows 8..15 -> 0)
        const v8f C1 = { lo ? dx  : 0.0f, lo ? dy  : 0.0f, lo ? dz  : 0.0f,
                         0.0f, 0.0f, 0.0f, 0.0f, 0.0f };
        const v8f C2 = { lo ? dxp : 0.0f, lo ? dyp : 0.0f, lo ? dzp : 0.0f,
                         0.0f, 0.0f, 0.0f, 0.0f, 0.0f };

        // ---- D = cell^T * offsets^T + diff : full dist_vec per pair ----
        const v8f D1 = __builtin_amdgcn_wmma_f32_16x16x4_f32(
            false, Am, false, B1, (short)0, C1, false, false);
        const v8f D2 = __builtin_amdgcn_wmma_f32_16x16x4_f32(
            false, Am, false, B2, (short)0, C2, false, false);

        // lane n (0..15): D1 -> pair pbase+n, D2 -> pair pbase+16+n
        // raw v_sqrt_f32 (~1 ulp) instead of the ~30-instruction IEEE sqrt expansion
        const float r1 = __builtin_amdgcn_sqrtf(D1[0] * D1[0] + D1[1] * D1[1] + D1[2] * D1[2]);
        const float r2 = __builtin_amdgcn_sqrtf(D2[0] * D2[0] + D2[1] * D2[1] + D2[2] * D2[2]);

        // route pair pbase+t back to lane t, apply mask, coalesced NT store
        const float r2s = __shfl(r2, t - 16, 32);
        float r = lo ? r1 : r2s;
        r = mk ? r : 0.0f;
        __builtin_nontemporal_store(r, &out[gp]);
    }
}

extern "C" void kernel_launch(void* const* d_in, const int* in_sizes, int n_in,
                              void* d_out, int out_size, void* d_ws, size_t ws_size,
                              hipStream_t stream) {
    const float*          positions    = (const float*)d_in[0];
    const long long*      neighbors    = (const long long*)d_in[1];
    const float*          cell         = (const float*)d_in[2];
    const float*          cell_offsets = (const float*)d_in[3];
    const unsigned char*  neighbor_mask= (const unsigned char*)d_in[4];
    float*                out          = (float*)d_out;

    // 2048 blocks x 256 threads = 16384 waves -> 16 chunk-iterations per wave
    AtomDistances_kernel<<<2048, 256, 0, stream>>>(
        positions, neighbors, cell, cell_offsets, neighbor_mask, out);
}